// HybridGCN_64836826300661
// MI455X (gfx1250) — compile-verified
//
#include <hip/hip_runtime.h>
#include <hip/hip_bf16.h>
#include <math.h>

#define NN   100000   // nodes
#define NE   1200000  // edges
#define NH   10000    // hyperedges
#define NI   1200000  // incidences
#define DIMF 64
#define NCLS 40

typedef __attribute__((ext_vector_type(2))) float v2f;
typedef __attribute__((ext_vector_type(8))) float v8f;

static __device__ __forceinline__ void atomic_add_f(float* p, float v) {
  // native global_atomic_add_f32 (no CAS loop)
  unsafeAtomicAdd(p, v);
}

// ---------------- degree computation ----------------
__global__ void k_deg_init(float* deg, float* Dd, float* Bd) {
  int i = blockIdx.x * blockDim.x + threadIdx.x;
  if (i < NN) { deg[i] = 1.0f; Dd[i] = 0.0f; }  // self-loop contributes 1
  if (i < NH) { Bd[i] = 0.0f; }
}

__global__ void k_deg_count(const int* __restrict__ ei, const int* __restrict__ hnd,
                            const int* __restrict__ hid,
                            float* deg, float* Dd, float* Bd) {
  int e = blockIdx.x * blockDim.x + threadIdx.x;
  if (e >= NE) return;
  atomic_add_f(&deg[ei[NE + e]], 1.0f);  // dst in-degree
  atomic_add_f(&Dd[hnd[e]], 1.0f);       // node hyper-degree
  atomic_add_f(&Bd[hid[e]], 1.0f);       // hyperedge degree
}

__global__ void k_deg_finish(const float* deg, float* dinv, float* Dd, float* Bd) {
  int i = blockIdx.x * blockDim.x + threadIdx.x;
  if (i < NN) {
    float d = deg[i];
    dinv[i] = (d > 0.0f) ? rsqrtf(d) : 0.0f;
    float dd = Dd[i];
    Dd[i] = (dd > 0.0f) ? 1.0f / dd : 0.0f;
  }
  if (i < NH) {
    float b = Bd[i];
    Bd[i] = (b > 0.0f) ? 1.0f / b : 0.0f;
  }
}

// ---------------- fp32 WMMA GEMM: out[rows x NCOLS] = X[rows x 64] * W[64 x NCOLS] ----------------
template <int NCOLS>
__global__ void __launch_bounds__(256)
k_gemm_wmma(const float* __restrict__ X, const float* __restrict__ W,
            float* __restrict__ out, int rows) {
  __shared__ float ldsW[64 * 64];
  const int tid = threadIdx.x;
  for (int i = tid; i < 64 * 64; i += 256) {
    int kk = i >> 6, n = i & 63;
    ldsW[i] = (n < NCOLS) ? W[kk * NCOLS + n] : 0.0f;
  }
  __syncthreads();

  const int lane = tid & 31;
  const int wv   = tid >> 5;
  const int tile = blockIdx.x * 8 + wv;
  const int m0   = tile * 16;
  if (m0 >= rows) return;   // wave-uniform exit -> EXEC all-ones inside

  constexpr int NT = (NCOLS + 15) / 16;
  v8f acc[NT];
#pragma unroll
  for (int t = 0; t < NT; ++t)
#pragma unroll
    for (int r = 0; r < 8; ++r) acc[t][r] = 0.0f;

  const int arow = m0 + (lane & 15);
  const int koff = (lane >> 4) << 1;  // A: lanes 0-15 hold K=0,1; lanes 16-31 hold K=2,3
  const int kk   = lane >> 4;         // B: lanes 0-15 row K, lanes 16-31 row K+1 (pair K+2 in v1)
  const int ncol = lane & 15;

#pragma unroll
  for (int k = 0; k < 64; k += 4) {
    v2f a = *(const v2f*)(X + (size_t)arow * 64 + k + koff);
#pragma unroll
    for (int t = 0; t < NT; ++t) {
      v2f b;
      b.x = ldsW[(k + kk) * 64 + t * 16 + ncol];
      b.y = ldsW[(k + kk + 2) * 64 + t * 16 + ncol];
      acc[t] = __builtin_amdgcn_wmma_f32_16x16x4_f32(
          false, a, false, b, (short)0, acc[t], false, false);
    }
  }

  const int rbase = m0 + ((lane >> 4) << 3);  // C/D: lanes 0-15 -> M=r, lanes 16-31 -> M=8+r
#pragma unroll
  for (int t = 0; t < NT; ++t) {
    int col = t * 16 + ncol;
    if (col < NCOLS) {
#pragma unroll
      for (int r = 0; r < 8; ++r)
        out[(size_t)(rbase + r) * NCOLS + col] = acc[t][r];
    }
  }
}

// ---------------- GCN: self-loop init, edge scatter, bias/act ----------------
__global__ void k_gcn_self(const float* __restrict__ h, const float* __restrict__ dinv,
                           float* __restrict__ agg, int C, int total) {
  int i = blockIdx.x * blockDim.x + threadIdx.x;
  if (i >= total) return;
  int row = i / C;
  float s = dinv[row];
  agg[i] = h[i] * s * s;
}

template <int HALFC>
__global__ void k_gcn_edges(const int* __restrict__ ei, const float* __restrict__ dinv,
                            const float* __restrict__ h, float* __restrict__ agg) {
  int t = blockIdx.x * blockDim.x + threadIdx.x;
  if (t >= NE * HALFC) return;
  int e = t / HALFC;
  int j = t - e * HALFC;
  int s = ei[e];
  int d = ei[NE + e];
  float coef = dinv[s] * dinv[d];
  const int C = 2 * HALFC;
  float2 v = *((const float2*)(h + (size_t)s * C) + j);
  float* ap = agg + (size_t)d * C + 2 * j;
  atomic_add_f(ap,     v.x * coef);
  atomic_add_f(ap + 1, v.y * coef);
}

__global__ void k_bias_act(float* __restrict__ y, const float* __restrict__ b,
                           int C, int relu, int total) {
  int i = blockIdx.x * blockDim.x + threadIdx.x;
  if (i >= total) return;
  int c = i % C;
  float v = y[i] + b[c];
  if (relu) v = fmaxf(v, 0.0f);
  y[i] = v;
}

// ---------------- Hypergraph: scatter to hyperedges, back to nodes, finalize ----------------
template <int HALFC>
__global__ void k_scat_he(const int* __restrict__ hnd, const int* __restrict__ hid,
                          const float* __restrict__ h, float* __restrict__ m) {
  int t = blockIdx.x * blockDim.x + threadIdx.x;
  if (t >= NI * HALFC) return;
  int e = t / HALFC;
  int j = t - e * HALFC;
  const int C = 2 * HALFC;
  int node = hnd[e], he = hid[e];
  float2 v = *((const float2*)(h + (size_t)node * C) + j);
  float* mp = m + (size_t)he * C + 2 * j;
  atomic_add_f(mp,     v.x);
  atomic_add_f(mp + 1, v.y);
}

template <int HALFC>
__global__ void k_scat_node(const int* __restrict__ hnd, const int* __restrict__ hid,
                            const float* __restrict__ m, const float* __restrict__ Binv,
                            float* __restrict__ out) {
  int t = blockIdx.x * blockDim.x + threadIdx.x;
  if (t >= NI * HALFC) return;
  int e = t / HALFC;
  int j = t - e * HALFC;
  const int C = 2 * HALFC;
  int node = hnd[e], he = hid[e];
  float bi = Binv[he];
  float2 v = *((const float2*)(m + (size_t)he * C) + j);
  float* op = out + (size_t)node * C + 2 * j;
  atomic_add_f(op,     v.x * bi);
  atomic_add_f(op + 1, v.y * bi);
}

__global__ void k_hyp_fin(float* __restrict__ y, const float* __restrict__ Dinv,
                          const float* __restrict__ b, int C, int relu, int total) {
  int i = blockIdx.x * blockDim.x + threadIdx.x;
  if (i >= total) return;
  int row = i / C;
  int c = i - row * C;
  float v = y[i] * Dinv[row] + b[c];
  if (relu) v = fmaxf(v, 0.0f);
  y[i] = v;
}

// ---------------- per-node 2-way softmax gate (wave per node) ----------------
__global__ void __launch_bounds__(256)
k_attention(const float* __restrict__ xa, const float* __restrict__ xb,
            const float* __restrict__ w, float* __restrict__ out, int C) {
  int lane = threadIdx.x & 31;
  int node = blockIdx.x * 8 + (threadIdx.x >> 5);
  if (node >= NN) return;
  float s0 = 0.0f, s1 = 0.0f;
  for (int c = lane; c < C; c += 32) {
    s0 += xa[(size_t)node * C + c] * w[2 * c];
    s1 += xb[(size_t)node * C + c] * w[2 * c + 1];
  }
  for (int m = 16; m; m >>= 1) {
    s0 += __shfl_xor(s0, m, 32);
    s1 += __shfl_xor(s1, m, 32);
  }
  float mx = fmaxf(s0, s1);
  float e0 = expf(s0 - mx), e1 = expf(s1 - mx);
  float inv = 1.0f / (e0 + e1);
  float a0 = e0 * inv, a1 = e1 * inv;
  for (int c = lane; c < C; c += 32) {
    size_t o = (size_t)node * C + c;
    out[o] = a0 * xa[o] + a1 * xb[o];
  }
}

// ---------------- fused final attention + log_softmax (C == 40) ----------------
__global__ void __launch_bounds__(256)
k_attn_lsm(const float* __restrict__ xa, const float* __restrict__ xb,
           const float* __restrict__ w, float* __restrict__ out) {
  const int C = NCLS;
  int lane = threadIdx.x & 31;
  int node = blockIdx.x * 8 + (threadIdx.x >> 5);
  if (node >= NN) return;
  bool has2 = lane < (C - 32);
  size_t base = (size_t)node * C;
  float a0_ = xa[base + lane], b0_ = xb[base + lane];
  float s0 = a0_ * w[2 * lane];
  float s1 = b0_ * w[2 * lane + 1];
  float a1_ = 0.0f, b1_ = 0.0f;
  if (has2) {
    a1_ = xa[base + lane + 32];
    b1_ = xb[base + lane + 32];
    s0 += a1_ * w[2 * (lane + 32)];
    s1 += b1_ * w[2 * (lane + 32) + 1];
  }
  for (int m = 16; m; m >>= 1) {
    s0 += __shfl_xor(s0, m, 32);
    s1 += __shfl_xor(s1, m, 32);
  }
  float mxs = fmaxf(s0, s1);
  float e0 = expf(s0 - mxs), e1 = expf(s1 - mxs);
  float inv = 1.0f / (e0 + e1);
  float g0 = e0 * inv, g1 = e1 * inv;
  float y0 = g0 * a0_ + g1 * b0_;
  float y1 = g0 * a1_ + g1 * b1_;   // only valid when has2
  // row max
  float mx = y0;
  if (has2) mx = fmaxf(mx, y1);
  for (int m = 16; m; m >>= 1) mx = fmaxf(mx, __shfl_xor(mx, m, 32));
  // sum exp
  float se = expf(y0 - mx) + (has2 ? expf(y1 - mx) : 0.0f);
  for (int m = 16; m; m >>= 1) se += __shfl_xor(se, m, 32);
  float lse = logf(se);
  out[base + lane] = y0 - mx - lse;
  if (has2) out[base + lane + 32] = y1 - mx - lse;
}

// ---------------- launch ----------------
extern "C" void kernel_launch(void* const* d_in, const int* in_sizes, int n_in,
                              void* d_out, int out_size, void* d_ws, size_t ws_size,
                              hipStream_t stream) {
  (void)in_sizes; (void)n_in; (void)out_size; (void)ws_size;
  const float* x   = (const float*)d_in[0];
  const int*   ei  = (const int*)d_in[1];
  const int*   hnd = (const int*)d_in[2];
  const int*   hid = (const int*)d_in[3];
  const float* W1g = (const float*)d_in[4];
  const float* b1g = (const float*)d_in[5];
  const float* W1h = (const float*)d_in[6];
  const float* b1h = (const float*)d_in[7];
  const float* aw1 = (const float*)d_in[8];
  const float* W2g = (const float*)d_in[9];
  const float* b2g = (const float*)d_in[10];
  const float* W2h = (const float*)d_in[11];
  const float* b2h = (const float*)d_in[12];
  const float* aw2 = (const float*)d_in[13];
  float* out = (float*)d_out;

  float* ws   = (float*)d_ws;
  float* deg  = ws;  ws += NN;
  float* dinv = ws;  ws += NN;
  float* Dd   = ws;  ws += NN;              // node hyper-degree -> 1/D
  float* Bd   = ws;  ws += NH;              // hyperedge degree -> 1/B
  float* bufA = ws;  ws += (size_t)NN * 64;
  float* bufB = ws;  ws += (size_t)NN * 64;
  float* bufC = ws;  ws += (size_t)NN * 64;
  float* bufM = ws;  ws += (size_t)NH * 64;
  float* bufE = ws;  ws += (size_t)NN * NCLS;

  const int B = 256;
  auto cdiv = [](long n, long d) { return (unsigned)((n + d - 1) / d); };
  const unsigned gemmGrid = cdiv(NN / 16, 8);

  // degrees / normalizers
  k_deg_init<<<cdiv(NN, B), B, 0, stream>>>(deg, Dd, Bd);
  k_deg_count<<<cdiv(NE, B), B, 0, stream>>>(ei, hnd, hid, deg, Dd, Bd);
  k_deg_finish<<<cdiv(NN, B), B, 0, stream>>>(deg, dinv, Dd, Bd);

  // ---- layer 1 GCN: x1 = relu(Ahat * (x W1g) + b1g) ----
  k_gemm_wmma<64><<<gemmGrid, 256, 0, stream>>>(x, W1g, bufA, NN);            // h1
  k_gcn_self<<<cdiv((long)NN * 64, B), B, 0, stream>>>(bufA, dinv, bufB, 64, NN * 64);
  k_gcn_edges<32><<<cdiv((long)NE * 32, B), B, 0, stream>>>(ei, dinv, bufA, bufB);
  k_bias_act<<<cdiv((long)NN * 64, B), B, 0, stream>>>(bufB, b1g, 64, 1, NN * 64);  // bufB = x1

  // ---- layer 1 hypergraph: x_hyper = relu(D^-1 H B^-1 H^T (x1 W1h) + b1h) ----
  k_gemm_wmma<64><<<gemmGrid, 256, 0, stream>>>(bufB, W1h, bufA, NN);         // h_hyp
  hipMemsetAsync(bufM, 0, (size_t)NH * 64 * sizeof(float), stream);
  k_scat_he<32><<<cdiv((long)NI * 32, B), B, 0, stream>>>(hnd, hid, bufA, bufM);
  hipMemsetAsync(bufC, 0, (size_t)NN * 64 * sizeof(float), stream);
  k_scat_node<32><<<cdiv((long)NI * 32, B), B, 0, stream>>>(hnd, hid, bufM, Bd, bufC);
  k_hyp_fin<<<cdiv((long)NN * 64, B), B, 0, stream>>>(bufC, Dd, b1h, 64, 1, NN * 64); // bufC = x_hyper

  // ---- branch attention 1: xa = gate(x1, x_hyper) ----
  k_attention<<<cdiv(NN, 8), 256, 0, stream>>>(bufB, bufC, aw1, bufA, 64);    // bufA = xa

  // ---- layer 2 GCN: x2 = Ahat * (xa W2g) + b2g ----
  k_gemm_wmma<NCLS><<<gemmGrid, 256, 0, stream>>>(bufA, W2g, bufB, NN);       // bufB = h2 (stride 40)
  k_gcn_self<<<cdiv((long)NN * NCLS, B), B, 0, stream>>>(bufB, dinv, bufA, NCLS, NN * NCLS);
  k_gcn_edges<20><<<cdiv((long)NE * 20, B), B, 0, stream>>>(ei, dinv, bufB, bufA);
  k_bias_act<<<cdiv((long)NN * NCLS, B), B, 0, stream>>>(bufA, b2g, NCLS, 0, NN * NCLS); // bufA = x2

  // ---- layer 2 hypergraph: xh2 = D^-1 H B^-1 H^T (x_hyper W2h) + b2h ----
  k_gemm_wmma<NCLS><<<gemmGrid, 256, 0, stream>>>(bufC, W2h, bufB, NN);       // bufB = h (stride 40)
  hipMemsetAsync(bufM, 0, (size_t)NH * NCLS * sizeof(float), stream);
  k_scat_he<20><<<cdiv((long)NI * 20, B), B, 0, stream>>>(hnd, hid, bufB, bufM);
  hipMemsetAsync(bufE, 0, (size_t)NN * NCLS * sizeof(float), stream);
  k_scat_node<20><<<cdiv((long)NI * 20, B), B, 0, stream>>>(hnd, hid, bufM, Bd, bufE);
  k_hyp_fin<<<cdiv((long)NN * NCLS, B), B, 0, stream>>>(bufE, Dd, b2h, NCLS, 0, NN * NCLS); // bufE = xh2

  // ---- branch attention 2 + log_softmax -> d_out ----
  k_attn_lsm<<<cdiv(NN, 8), 256, 0, stream>>>(bufA, bufE, aw2, out);
}